// Decoder_38912403702378
// MI455X (gfx1250) — compile-verified
//
#include <hip/hip_runtime.h>
#include <math.h>

// Problem constants (match reference)
#define BB   1024
#define TT   64
#define EE   512
#define DD   512

typedef float v2f __attribute__((ext_vector_type(2)));
typedef float v4f __attribute__((ext_vector_type(4)));
typedef float v8f __attribute__((ext_vector_type(8)));

// LDS geometry:
//  APITCH=36 (floats): row pitch 144B (16B aligned -> b128 staging stores);
//      fragment reads hit all 64 banks conflict-free (gcd(36,64)=4 -> 16 rows
//      x 4 cols cover all banks).
//  BP2=144 (float2 units): B tile stored k-pair-interleaved so a fragment is
//      one contiguous b64; 2*BP2 == 32 (mod 64) -> lane halves 0-15 / 16-31
//      read disjoint 32-bank halves; staging b64 stores cover all 64 banks.
#define APITCH 36
#define BP2    144

__device__ __forceinline__ float fast_tanh(float x) {
#if __has_builtin(__builtin_amdgcn_tanhf)
    return __builtin_amdgcn_tanhf(x);     // native V_TANH_F32
#else
    return tanhf(x);
#endif
}

// ---------------------------------------------------------------------------
// Generic WMMA-f32 GEMM:  C[m][n] = sum_k A[m][k] * W[n][k]
//                                  + bias[n] + bias2[n] + rowv[m]*colv[n]
// A row-major M x K, optionally split at K0 between A0/A1 (both ld=K0; used
// for the [h,c] concat; K0 is a multiple of 32 so a K-stage never straddles).
// W row-major N x K (computes A @ W^T).  C row-major M x N.
// Block = 256 threads (8 waves).  Workgroup tile 128(M) x 128(N), K step 32.
// Wave w: rows [ (w&3)*32, +32 ), cols [ (w>>2)*64, +64 )
//   -> 2 A-fragments x 4 B-fragments -> 8 WMMA per K-step of 4 (B reused 2x).
// ---------------------------------------------------------------------------
__global__ __launch_bounds__(256)
void wmma_gemm_nt(const float* __restrict__ A0, const float* __restrict__ A1,
                  int K0,
                  const float* __restrict__ W,
                  const float* __restrict__ bias,
                  const float* __restrict__ bias2,
                  const float* __restrict__ rowv,
                  const float* __restrict__ colv,
                  float* __restrict__ C, int M, int N, int K)
{
    __shared__ __align__(16) float Alds[128 * APITCH];  // 18.0 KB
    __shared__ v2f Blds2[16 * BP2];                     // 18.0 KB (k-pairs x n)

    const int tid  = threadIdx.x;
    const int wid  = tid >> 5;
    const int lane = tid & 31;
    const int half = lane >> 4;   // K-pair select for f32 WMMA fragments
    const int lrow = lane & 15;   // M (for A) / N (for B,C)
    const int wm   = wid & 3;     // which 32-row tile
    const int wn   = wid >> 2;    // which 64-col tile
    const int m0   = blockIdx.y * 128;
    const int n0   = blockIdx.x * 128;

    v8f acc[2][4];
    #pragma unroll
    for (int r = 0; r < 2; ++r)
        #pragma unroll
        for (int j = 0; j < 4; ++j)
            acc[r][j] = (v8f){0.f,0.f,0.f,0.f,0.f,0.f,0.f,0.f};

    for (int kt = 0; kt < K; kt += 32) {
        // ---- Stage A tile 128x32 as float4 (global_load_b128, coalesced;
        //      ds_store_b128: 16B-aligned since APITCH*4 and c4*4 are mult of 16)
        for (int i = tid; i < 128 * 8; i += 256) {
            int r  = i >> 3;
            int c4 = (i & 7) * 4;
            int gk = kt + c4;                     // whole float4 on one side of K0
            const float* src = (gk < K0)
                ? &A0[(size_t)(m0 + r) * K0 + gk]
                : &A1[(size_t)(m0 + r) * K0 + (gk - K0)];
            v4f v = *(const v4f*)src;
            *(v4f*)&Alds[r * APITCH + c4] = v;
        }
        // ---- Stage W tile 32x128 k-pair-interleaved:
        //      Blds2[p][n] = (W[n0+n][kt+2p], W[n0+n][kt+2p+1])
        //      n-fast mapping: conflict-free b64 stores; weights are L2-resident.
        for (int i = tid; i < 8 * 128; i += 256) {
            int n  = i & 127;
            int k4 = (i >> 7) * 4;
            v4f v = *(const v4f*)&W[(size_t)(n0 + n) * K + (kt + k4)];
            int p0 = k4 >> 1;
            v2f lo; lo.x = v.x; lo.y = v.y;
            v2f hi; hi.x = v.z; hi.y = v.w;
            Blds2[(p0 + 0) * BP2 + n] = lo;
            Blds2[(p0 + 1) * BP2 + n] = hi;
        }
        __syncthreads();

        #pragma unroll
        for (int kk = 0; kk < 32; kk += 4) {
            // A fragments 16x4: lanes 0-15 hold K={kk,kk+1}, lanes 16-31 K={kk+2,kk+3}
            const float* ap0 = &Alds[(wm * 32 + lrow)      * APITCH + kk + 2 * half];
            const float* ap1 = &Alds[(wm * 32 + 16 + lrow) * APITCH + kk + 2 * half];
            v2f a0; a0.x = ap0[0]; a0.y = ap0[1];
            v2f a1; a1.x = ap1[0]; a1.y = ap1[1];
            const int p = (kk >> 1) + half;       // k-pair index for this lane half
            #pragma unroll
            for (int j = 0; j < 4; ++j) {
                int col = wn * 64 + j * 16 + lrow;
                v2f bb = Blds2[p * BP2 + col];    // single ds_load_b64, no shuffles
                acc[0][j] = __builtin_amdgcn_wmma_f32_16x16x4_f32(
                    false, a0, false, bb, (short)0, acc[0][j], false, false);
                acc[1][j] = __builtin_amdgcn_wmma_f32_16x16x4_f32(
                    false, a1, false, bb, (short)0, acc[1][j], false, false);
            }
        }
        __syncthreads();
    }

    // Epilogue: bias + optional rank-1 (rowv[m]*colv[n]) term
    #pragma unroll
    for (int j = 0; j < 4; ++j) {
        int ncol = n0 + wn * 64 + j * 16 + lrow;
        float badd = 0.0f;
        if (bias)  badd += bias[ncol];
        if (bias2) badd += bias2[ncol];
        float cv = colv ? colv[ncol] : 0.0f;
        #pragma unroll
        for (int r = 0; r < 2; ++r) {
            #pragma unroll
            for (int v = 0; v < 8; ++v) {
                int grow = m0 + wm * 32 + r * 16 + v + 8 * half; // C/D: M = v + 8*half
                float res = acc[r][j][v] + badd;
                if (rowv) res += rowv[grow] * cv;
                C[(size_t)grow * N + ncol] = res;
            }
        }
    }
}

// ---------------------------------------------------------------------------
// Fused attention step (one workgroup per batch element b):
//   scores[t] = b_a3 + sum_e tanh(z1[b,e] + z2[b,t,e]) * W_a3[e]
//   attn      = softmax_t(scores)
//   ctx[b,e]  = sum_t attn[t] * X[b,t,e]
//   ytilde[b] = b_fc + sum_e ctx[b,e]*W_fc[e] + W_fc[E]*y_hist[b,step]
// z2 (134 MB) streams through L2 (192 MB) -> near-L2-bandwidth phase.
// ---------------------------------------------------------------------------
__global__ __launch_bounds__(256)
void attn_fused(const float* __restrict__ z1, const float* __restrict__ z2,
                const float* __restrict__ X,
                const float* __restrict__ Wa3, const float* __restrict__ ba3,
                const float* __restrict__ Wfc, const float* __restrict__ bfc,
                const float* __restrict__ yhist, int step,
                float* __restrict__ ctx, float* __restrict__ ytilde)
{
    __shared__ float z1s[EE];
    __shared__ float wa3s[EE];
    __shared__ float sc[TT];
    __shared__ float wred[8];

    const int b    = blockIdx.x;
    const int tid  = threadIdx.x;
    const int wid  = tid >> 5;
    const int lane = tid & 31;

    for (int i = tid; i < EE; i += 256) {
        z1s[i]  = z1[(size_t)b * EE + i];
        wa3s[i] = Wa3[i];
    }
    __syncthreads();

    // Scores: each wave owns 8 of the 64 time rows; lane-strided e-reduction.
    for (int t = wid; t < TT; t += 8) {
        const float* z2row = z2 + ((size_t)b * TT + t) * EE;
        if (t + 8 < TT)   // hint next row into cache (global_prefetch_b8)
            __builtin_prefetch(z2 + ((size_t)b * TT + t + 8) * EE + lane * 16, 0, 0);
        float s = 0.0f;
        for (int e = lane; e < EE; e += 32)
            s += fast_tanh(z1s[e] + z2row[e]) * wa3s[e];
        #pragma unroll
        for (int off = 16; off > 0; off >>= 1) s += __shfl_xor(s, off, 32);
        if (lane == 0) sc[t] = s + ba3[0];
    }
    __syncthreads();

    // Softmax over 64 scores: wave 0, two values per lane.
    if (tid < 32) {
        float a0 = sc[tid], a1 = sc[tid + 32];
        float m = fmaxf(a0, a1);
        #pragma unroll
        for (int off = 16; off > 0; off >>= 1) m = fmaxf(m, __shfl_xor(m, off, 32));
        float e0 = __expf(a0 - m), e1 = __expf(a1 - m);
        float s = e0 + e1;
        #pragma unroll
        for (int off = 16; off > 0; off >>= 1) s += __shfl_xor(s, off, 32);
        float inv = 1.0f / s;
        sc[tid] = e0 * inv;
        sc[tid + 32] = e1 * inv;
    }
    __syncthreads();

    // Context + partial dot with W_fc
    float part = 0.0f;
    for (int e = tid; e < EE; e += 256) {
        const float* xb = X + (size_t)b * TT * EE + e;
        float a = 0.0f;
        #pragma unroll 8
        for (int t = 0; t < TT; ++t) a += sc[t] * xb[(size_t)t * EE];
        ctx[(size_t)b * EE + e] = a;
        part += a * Wfc[e];
    }
    #pragma unroll
    for (int off = 16; off > 0; off >>= 1) part += __shfl_xor(part, off, 32);
    if (lane == 0) wred[wid] = part;
    __syncthreads();
    if (tid == 0) {
        float tot = 0.0f;
        #pragma unroll
        for (int w = 0; w < 8; ++w) tot += wred[w];
        ytilde[b] = tot + Wfc[EE] * yhist[(size_t)b * TT + step] + bfc[0];
    }
}

// ---------------------------------------------------------------------------
// LSTM pointwise update (torch gate order i,f,g,o).
// ---------------------------------------------------------------------------
__device__ __forceinline__ float sigm(float x) { return 1.0f / (1.0f + __expf(-x)); }

__global__ __launch_bounds__(256)
void lstm_update(const float* __restrict__ gates, const float* __restrict__ c_in,
                 float* __restrict__ h_out, float* __restrict__ c_out)
{
    int idx = blockIdx.x * 256 + threadIdx.x;          // 0 .. B*D-1
    int b = idx >> 9;                                  // D = 512
    int j = idx & 511;
    const float* g = gates + (size_t)b * (4 * DD);
    float ig = g[j];
    float fg = g[DD + j];
    float gg = g[2 * DD + j];
    float og = g[3 * DD + j];
    float c  = sigm(fg) * c_in[idx] + sigm(ig) * fast_tanh(gg);
    h_out[idx] = sigm(og) * fast_tanh(c);
    c_out[idx] = c;
}

// ---------------------------------------------------------------------------
// Final projection: out[b] = b_ff + [h, ctx] . W_ff
// ---------------------------------------------------------------------------
__global__ __launch_bounds__(256)
void final_proj(const float* __restrict__ h, const float* __restrict__ ctx,
                const float* __restrict__ Wff, const float* __restrict__ bff,
                float* __restrict__ out)
{
    __shared__ float wred[8];
    const int b = blockIdx.x, tid = threadIdx.x;
    const int wid = tid >> 5, lane = tid & 31;
    float p = 0.0f;
    for (int i = tid; i < DD; i += 256) {
        p += h[(size_t)b * DD + i]   * Wff[i];
        p += ctx[(size_t)b * EE + i] * Wff[DD + i];
    }
    #pragma unroll
    for (int off = 16; off > 0; off >>= 1) p += __shfl_xor(p, off, 32);
    if (lane == 0) wred[wid] = p;
    __syncthreads();
    if (tid == 0) {
        float tot = 0.0f;
        #pragma unroll
        for (int w = 0; w < 8; ++w) tot += wred[w];
        out[b] = tot + bff[0];
    }
}

// ---------------------------------------------------------------------------
extern "C" void kernel_launch(void* const* d_in, const int* in_sizes, int n_in,
                              void* d_out, int out_size, void* d_ws, size_t ws_size,
                              hipStream_t stream)
{
    (void)in_sizes; (void)n_in; (void)out_size; (void)ws_size;

    const float* X   = (const float*)d_in[0];   // (B, T, E)
    const float* Yh  = (const float*)d_in[1];   // (B, T, 1)
    const float* Wa1 = (const float*)d_in[2];   // (E, 2D)
    const float* ba1 = (const float*)d_in[3];   // (E,)
    const float* Wa2 = (const float*)d_in[4];   // (E, E)
    const float* ba2 = (const float*)d_in[5];   // (E,)
    const float* Wa3 = (const float*)d_in[6];   // (1, E)
    const float* ba3 = (const float*)d_in[7];   // (1,)
    const float* Wih = (const float*)d_in[8];   // (4D, 1)
    const float* Whh = (const float*)d_in[9];   // (4D, D)
    const float* bih = (const float*)d_in[10];  // (4D,)
    const float* bhh = (const float*)d_in[11];  // (4D,)
    const float* Wfc = (const float*)d_in[12];  // (1, E+1)
    const float* bfc = (const float*)d_in[13];  // (1,)
    const float* Wff = (const float*)d_in[14];  // (1, D+E)
    const float* bff = (const float*)d_in[15];  // (1,)
    float* out = (float*)d_out;

    // Workspace layout (floats)
    float* ws    = (float*)d_ws;
    float* z2    = ws;                                   // B*T*E   = 33,554,432
    float* z1b   = z2    + (size_t)BB * TT * EE;         // B*E     =    524,288
    float* ctx   = z1b   + (size_t)BB * EE;              // B*E
    float* gates = ctx   + (size_t)BB * EE;              // B*4D    =  2,097,152
    float* h0    = gates + (size_t)BB * 4 * DD;          // B*D
    float* c0    = h0    + (size_t)BB * DD;
    float* h1    = c0    + (size_t)BB * DD;
    float* c1    = h1    + (size_t)BB * DD;
    float* ytil  = c1    + (size_t)BB * DD;              // B

    // h, c start at zero every call (ws is not re-poisoned between replays)
    hipMemsetAsync(h0, 0, (size_t)BB * DD * sizeof(float), stream);
    hipMemsetAsync(c0, 0, (size_t)BB * DD * sizeof(float), stream);

    // Loop-invariant: z2 = X @ W_a2^T + b_a2   (M=B*T=65536, N=E, K=E)
    wmma_gemm_nt<<<dim3(EE / 128, (BB * TT) / 128), 256, 0, stream>>>(
        X, X, EE, Wa2, ba2, nullptr, nullptr, nullptr,
        z2, BB * TT, EE, EE);

    float* h_cur = h0; float* c_cur = c0;
    float* h_nxt = h1; float* c_nxt = c1;

    for (int s = 0; s < TT; ++s) {
        // z1 = [h,c] @ W_a1^T + b_a1   (M=B, N=E, K=2D, split-A at K0=D)
        wmma_gemm_nt<<<dim3(EE / 128, BB / 128), 256, 0, stream>>>(
            h_cur, c_cur, DD, Wa1, ba1, nullptr, nullptr, nullptr,
            z1b, BB, EE, 2 * DD);

        // Fused attention + y_tilde
        attn_fused<<<BB, 256, 0, stream>>>(
            z1b, z2, X, Wa3, ba3, Wfc, bfc, Yh, s, ctx, ytil);

        // gates = h @ W_hh^T + (b_ih+b_hh) + ytil[m]*W_ih[n]   (M=B, N=4D, K=D)
        wmma_gemm_nt<<<dim3((4 * DD) / 128, BB / 128), 256, 0, stream>>>(
            h_cur, h_cur, DD, Whh, bih, bhh, ytil, Wih,
            gates, BB, 4 * DD, DD);

        // LSTM pointwise
        lstm_update<<<(BB * DD) / 256, 256, 0, stream>>>(
            gates, c_cur, h_nxt, c_nxt);

        float* th = h_cur; h_cur = h_nxt; h_nxt = th;
        float* tc = c_cur; c_cur = c_nxt; c_nxt = tc;
    }

    // out = [h, ctx] @ W_ff^T + b_ff
    final_proj<<<BB, 256, 0, stream>>>(h_cur, ctx, Wff, bff, out);
}